// MAGNN_metapath_specific_90666759619114
// MI455X (gfx1250) — compile-verified
//
#include <hip/hip_runtime.h>

// ---------------------------------------------------------------------------
// MAGNN metapath-specific aggregation for MI455X (gfx1250, wave32).
//
// Memory-bound problem (~1.8 GB total traffic @ 23.3 TB/s HBM, FLOPs trivial),
// so everything stays f32. The logits GEMM [E,32]x[32,8] is mapped onto
// V_WMMA_F32_16X16X4_F32 (16 edges x 8 heads per wave, 8 chained K=4 steps),
// which is free on the matrix pipe and keeps full f32 precision.
// ---------------------------------------------------------------------------

typedef __attribute__((ext_vector_type(2))) float v2f;
typedef __attribute__((ext_vector_type(8))) float v8f;

#define E_EDGES 1000000
#define N_NODES 100000
#define H_HEADS 8
#define D_DIM   32
#define LRELU_ALPHA 0.01f
#define LDS_ROW_STRIDE 34   // 32 floats + 2 pad: keeps rows 8B-aligned (even) for b64 reads

// Monotonic float -> uint encoding so integer atomicMax == float max
// (deterministic, no CAS loops). enc(-inf) = 0x007FFFFF.
__device__ __forceinline__ unsigned enc_f32(float x) {
    unsigned u = __float_as_uint(x);
    return (u & 0x80000000u) ? ~u : (u | 0x80000000u);
}
__device__ __forceinline__ float dec_f32(unsigned k) {
    unsigned u = (k & 0x80000000u) ? (k ^ 0x80000000u) : ~k;
    return __uint_as_float(u);
}
#define ENC_NEG_INF 0x007FFFFFu

// ---------------------------------------------------------------------------
// K0: zero the output, init per-(node,head) running max (enc(-inf)) and sum.
// ---------------------------------------------------------------------------
__global__ void k0_init(float* __restrict__ out, unsigned* __restrict__ nodeMaxU,
                        float* __restrict__ nodeSum, int outElems, int nh) {
    int i = blockIdx.x * blockDim.x + threadIdx.x;
    if (i < outElems) out[i] = 0.0f;
    if (i < nh) {
        nodeMaxU[i] = ENC_NEG_INF;
        nodeSum[i]  = 0.0f;
    }
}

// ---------------------------------------------------------------------------
// K1: per wave: 16-edge tile.
//   Phase A (lane = feature dim d): L2-normalize feat0/1/2 via wave butterfly
//     reductions, hidden = (n0+n1+n2 + r0 + 2*r1 + 2*r2)/3; store to global
//     and stage the 16x32 tile in LDS.
//   Phase B: 8x chained V_WMMA_F32_16X16X4_F32 computing the 16x8 logits
//     block A[16x32] * attn^T[32x8 padded to 16 cols]; leaky-relu; store
//     logits and atomicMax into the segment max.
// ---------------------------------------------------------------------------
__global__ __launch_bounds__(256) void k1_hidden_logits(
    const float* __restrict__ feat0, const float* __restrict__ feat1,
    const float* __restrict__ feat2, const float* __restrict__ rvec,
    const float* __restrict__ attn,  const int* __restrict__ dst,
    float* __restrict__ hiddenG, float* __restrict__ logits,
    unsigned* __restrict__ nodeMaxU, int numTiles)
{
    __shared__ __align__(16) float lds[8 * 16 * LDS_ROW_STRIDE]; // 8 waves/block
    const int lane      = threadIdx.x & 31;
    const int waveInBlk = threadIdx.x >> 5;
    const int tile      = blockIdx.x * 8 + waveInBlk;
    if (tile >= numTiles) return;          // wave-uniform guard: EXEC full below
    float* tileLds = &lds[waveInBlk * 16 * LDS_ROW_STRIDE];

    // reference: feats[0]+=r0+r1+r2, feats[1]+=r1+r2 => hidden gets (r0+2r1+2r2)/3
    const float rthird = (rvec[lane] + 2.0f * rvec[D_DIM + lane]
                                     + 2.0f * rvec[2 * D_DIM + lane]) * (1.0f / 3.0f);

    const int base = tile * 16;

    // ---- Phase A: hidden for 16 edges, lane = d ----
    for (int i = 0; i < 16; ++i) {
        const int idx = (base + i) * D_DIM + lane;
        float f0 = feat0[idx], f1 = feat1[idx], f2 = feat2[idx];
        float s0 = f0 * f0, s1 = f1 * f1, s2 = f2 * f2;
        #pragma unroll
        for (int off = 16; off > 0; off >>= 1) {   // wave32 butterfly reduction
            s0 += __shfl_xor(s0, off, 32);
            s1 += __shfl_xor(s1, off, 32);
            s2 += __shfl_xor(s2, off, 32);
        }
        const float i0 = 1.0f / fmaxf(sqrtf(s0), 1e-12f);
        const float i1 = 1.0f / fmaxf(sqrtf(s1), 1e-12f);
        const float i2 = 1.0f / fmaxf(sqrtf(s2), 1e-12f);
        const float hid = (f0 * i0 + f1 * i1 + f2 * i2) * (1.0f / 3.0f) + rthird;
        hiddenG[idx] = hid;                        // reused by K3
        tileLds[i * LDS_ROW_STRIDE + lane] = hid;  // staged for WMMA A layout
    }
    // Within-wave LDS store->load: DS ops from one wave are in-order; the
    // compiler inserts the s_wait_dscnt for the dependent loads below.

    // ---- Phase B: B-matrix (attn^T, 4x16 per step) preload, branchless ----
    // f32 WMMA B layout (4x16): VGPR0 = rows K=0 (lanes 0-15) / K=2 (lanes
    // 16-31); VGPR1 = rows K=1 / K=3 -- mirrors the documented A layout.
    const int   n         = lane & 15;             // output column = head id
    const int   khalf     = (lane >> 4) * 2;       // 0 for lanes 0-15, 2 for 16-31
    const float headMask  = (n < H_HEADS) ? 1.0f : 0.0f;
    const int   nClamped  = n & 7;
    v2f bv[8];
    #pragma unroll
    for (int j = 0; j < 8; ++j) {
        bv[j].x = attn[nClamped * D_DIM + 4 * j + khalf]     * headMask;
        bv[j].y = attn[nClamped * D_DIM + 4 * j + khalf + 1] * headMask;
    }

    // ---- 8 chained f32 WMMAs over K=0..31 ----
    // A layout (16x4): lanes 0-15 hold row M=lane, K=(4j,4j+1); lanes 16-31
    // hold row M=lane-16, K=(4j+2,4j+3)  -> one 8B-aligned ds_load_b64/lane.
    v8f acc = {0.f, 0.f, 0.f, 0.f, 0.f, 0.f, 0.f, 0.f};
    #pragma unroll
    for (int j = 0; j < 8; ++j) {
        const v2f av = *(const v2f*)&tileLds[(lane & 15) * LDS_ROW_STRIDE + 4 * j + khalf];
        acc = __builtin_amdgcn_wmma_f32_16x16x4_f32(
            /*neg_a=*/false, av, /*neg_b=*/false, bv[j],
            /*c_mod=*/(short)0, acc, /*reuse_a=*/false, /*reuse_b=*/false);
    }

    // ---- C/D layout: VGPR i -> M = i + (lane>=16 ? 8 : 0), N = lane&15 ----
    if (n < H_HEADS) {
        const int h     = n;
        const int mbase = base + ((lane >> 4) << 3);   // +0 or +8
        #pragma unroll
        for (int i = 0; i < 8; ++i) {
            float a = acc[i];
            a = (a > 0.0f) ? a : LRELU_ALPHA * a;      // leaky relu
            const int e = mbase + i;
            logits[e * H_HEADS + h] = a;
            atomicMax(&nodeMaxU[(unsigned)dst[e] * H_HEADS + h], enc_f32(a));
        }
    }
}

// ---------------------------------------------------------------------------
// K2: ex = exp(a - segmax), in place over logits; atomicAdd into node sums.
// ---------------------------------------------------------------------------
__global__ void k2_expsum(float* logits, const int* __restrict__ dst,
                          const unsigned* __restrict__ nodeMaxU,
                          float* __restrict__ nodeSum)
{
    const int idx = blockIdx.x * blockDim.x + threadIdx.x;
    if (idx >= E_EDGES * H_HEADS) return;
    const int e = idx >> 3, h = idx & 7;
    const int nh = dst[e] * H_HEADS + h;
    const float m  = dec_f32(nodeMaxU[nh]);
    const float ex = __expf(logits[idx] - m);
    logits[idx] = ex;                       // overwrite logits with exp values
    atomicAdd(&nodeSum[nh], ex);
}

// ---------------------------------------------------------------------------
// K3: one wave per edge (lane = d): out[dst,h,:] += (ex/sum) * hidden[e,:]
//     8 fully-coalesced rows of global_atomic_add_f32 per edge.
// ---------------------------------------------------------------------------
__global__ __launch_bounds__(256) void k3_aggregate(
    const float* __restrict__ hiddenG, const float* __restrict__ exbuf,
    const float* __restrict__ nodeSum, const int* __restrict__ dst,
    float* __restrict__ out)
{
    const int wave = (int)((blockIdx.x * blockDim.x + threadIdx.x) >> 5);
    const int lane = threadIdx.x & 31;
    if (wave >= E_EDGES) return;
    const int e    = wave;
    const float hid = hiddenG[e * D_DIM + lane];
    const int node  = dst[e];
    const float* exRow  = &exbuf[e * H_HEADS];
    const float* sumRow = &nodeSum[node * H_HEADS];
    float* outRow = &out[(size_t)node * H_HEADS * D_DIM + lane];
    #pragma unroll
    for (int h = 0; h < H_HEADS; ++h) {
        const float att = exRow[h] / sumRow[h];   // uniform across wave -> bcast
        atomicAdd(&outRow[h * D_DIM], att * hid);
    }
}

// ---------------------------------------------------------------------------
extern "C" void kernel_launch(void* const* d_in, const int* in_sizes, int n_in,
                              void* d_out, int out_size, void* d_ws, size_t ws_size,
                              hipStream_t stream)
{
    const float* feat0 = (const float*)d_in[0];
    const float* feat1 = (const float*)d_in[1];
    const float* feat2 = (const float*)d_in[2];
    const float* rvec  = (const float*)d_in[3];
    const float* attn  = (const float*)d_in[4];
    const int*   dst   = (const int*)d_in[5];
    // d_in[6] = num_nodes scalar (compile-time N_NODES here)

    // Workspace layout: hidden[E,32] | logits/ex[E,8] | nodeMax[N,8] | nodeSum[N,8]
    char* ws = (char*)d_ws;
    float*    hiddenG  = (float*)ws;    ws += (size_t)E_EDGES * D_DIM   * sizeof(float);
    float*    logits   = (float*)ws;    ws += (size_t)E_EDGES * H_HEADS * sizeof(float);
    unsigned* nodeMaxU = (unsigned*)ws; ws += (size_t)N_NODES * H_HEADS * sizeof(unsigned);
    float*    nodeSum  = (float*)ws;

    float* out = (float*)d_out;
    const int outElems = N_NODES * H_HEADS * D_DIM;

    {   // K0
        const int blocks = (outElems + 255) / 256;
        k0_init<<<blocks, 256, 0, stream>>>(out, nodeMaxU, nodeSum,
                                            outElems, N_NODES * H_HEADS);
    }
    {   // K1: one wave per 16-edge tile, 8 waves/block
        const int numTiles = E_EDGES / 16;          // 62500, exact
        const int blocks   = (numTiles + 7) / 8;
        k1_hidden_logits<<<blocks, 256, 0, stream>>>(
            feat0, feat1, feat2, rvec, attn, dst,
            hiddenG, logits, nodeMaxU, numTiles);
    }
    {   // K2
        const int total = E_EDGES * H_HEADS;
        k2_expsum<<<(total + 255) / 256, 256, 0, stream>>>(logits, dst, nodeMaxU, nodeSum);
    }
    {   // K3: one wave per edge
        const long long threads = (long long)E_EDGES * 32;
        const int blocks = (int)((threads + 255) / 256);
        k3_aggregate<<<blocks, 256, 0, stream>>>(hiddenG, logits, nodeSum, dst, out);
    }
}